// ISCMembedding_89343909691671
// MI455X (gfx1250) — compile-verified
//
#include <hip/hip_runtime.h>
#include <math.h>

// ---- problem constants (match reference) ----
#define NB   4
#define NT   512
#define NF   257
#define NFP  272           // f padded to 17*16 for 16x16 WMMA tiles
#define NMIC 4
#define NP   10
#define NC   20            // 2*P conv input channels
#define ROWF (NC * NFP)    // 5440 floats per (b,t) feature row
#define NDM  128           // d_model
#define NK   100           // 2P * 5 (im2col K)
#define KSTEPS (NK / 4)    // 25 WMMA K-steps
#define NNT  17            // N tiles (f)
#define TPW  9             // N tiles per wave column (2 cols * 9 >= 17)
#define PAIRBLK (2 * NFP)  // 544 floats per K-pair block in xldB

typedef float v2f __attribute__((ext_vector_type(2)));
typedef float v8f __attribute__((ext_vector_type(8)));

__device__ __constant__ int IU0c[NP] = {0,0,0,0,1,1,1,2,2,3};
__device__ __constant__ int IU1c[NP] = {0,1,2,3,1,2,3,2,3,3};

// ---------------------------------------------------------------------------
// Kernel 1: per (b,f) time statistics: complex mean per mic, 1/x_norm.
// stats layout: [B*F][12] = {mean_r[4], mean_i[4], rnorm, pad...}
// ---------------------------------------------------------------------------
__global__ __launch_bounds__(128) void stats_kernel(const float* __restrict__ x,
                                                    float* __restrict__ stats) {
  const int b = blockIdx.x / NF, f = blockIdx.x % NF;
  const int tid = threadIdx.x;
  float sr[NMIC] = {0.f,0.f,0.f,0.f};
  float si[NMIC] = {0.f,0.f,0.f,0.f};
  float pw[NMIC] = {0.f,0.f,0.f,0.f};
  for (int it = 0; it < NT / 128; ++it) {
    const int t = tid + 128 * it;
    const float* xp = x + ((size_t)(b * NT + t) * NF + f) * (2 * NMIC);
    #pragma unroll
    for (int m = 0; m < NMIC; ++m) {
      const float r = xp[m], im = xp[NMIC + m];
      sr[m] += r; si[m] += im; pw[m] += r * r + im * im;
    }
  }
  __shared__ float red[128 * 12];
  #pragma unroll
  for (int m = 0; m < NMIC; ++m) {
    red[tid * 12 + m]     = sr[m];
    red[tid * 12 + 4 + m] = si[m];
    red[tid * 12 + 8 + m] = pw[m];
  }
  __syncthreads();
  for (int s = 64; s > 0; s >>= 1) {
    if (tid < s)
      for (int j = 0; j < 12; ++j) red[tid * 12 + j] += red[(tid + s) * 12 + j];
    __syncthreads();
  }
  if (tid == 0) {
    float* sp = stats + (size_t)(b * NF + f) * 12;
    const float invT = 1.0f / (float)NT;
    float tot = 0.f;
    #pragma unroll
    for (int m = 0; m < NMIC; ++m) {
      const float mr = red[m] * invT, mi = red[4 + m] * invT;
      tot += red[8 + m] * invT - mr * mr - mi * mi;   // E|x|^2 - |mean|^2
      sp[m] = mr; sp[4 + m] = mi;
    }
    tot = fmaxf(tot, 1e-10f);
    sp[8] = rsqrtf(tot);                              // 1 / x_norm
  }
}

// ---------------------------------------------------------------------------
// Kernel 2: per (b,t,f) SCM pair products + magnitude/phase warp -> 20 feats.
// Writes xs[b][t][c][f] with f padded to 272 (zeros) so the conv kernel's
// LDS fill reads coalesced 272-float rows.
// ---------------------------------------------------------------------------
__global__ __launch_bounds__(288) void feat_kernel(const float* __restrict__ x,
                                                   const float* __restrict__ stats,
                                                   const float* __restrict__ expo,
                                                   const float* __restrict__ ipd,
                                                   float* __restrict__ xs) {
  const int bt = blockIdx.x;           // b*T + t
  const int f  = threadIdx.x;
  if (f >= NFP) return;
  float* xrow = xs + (size_t)bt * ROWF;
  if (f >= NF) {                       // zero pad region (deterministic each call)
    #pragma unroll
    for (int c = 0; c < NC; ++c) xrow[c * NFP + f] = 0.f;
    return;
  }
  const int b = bt / NT;
  const float* xp = x + ((size_t)bt * NF + f) * (2 * NMIC);
  const float* sp = stats + (size_t)(b * NF + f) * 12;
  const float rn = sp[8];
  float zr[NMIC], zi[NMIC];
  #pragma unroll
  for (int m = 0; m < NMIC; ++m) {
    zr[m] = (xp[m] - sp[m]) * rn;
    zi[m] = (xp[NMIC + m] - sp[4 + m]) * rn;
  }
  const float se = 1.f / (1.f + __expf(-expo[f]));
  const float sI = 1.f / (1.f + __expf(-ipd[f]));
  #pragma unroll
  for (int p = 0; p < NP; ++p) {
    const int i = IU0c[p], j = IU1c[p];
    const float re = zr[i] * zr[j] + zi[i] * zi[j];   // z_i * conj(z_j)
    const float im = zi[i] * zr[j] - zr[i] * zi[j];
    const float mag = sqrtf(re * re + im * im);
    const float beta = __powf(mag, se);
    const float a2 = mag / (beta + 1e-10f);
    const float ang = atan2f(im, re) * sI;
    float sn, cs;
    __sincosf(ang, &sn, &cs);
    xrow[(2 * p)     * NFP + f] = a2 * cs;
    xrow[(2 * p + 1) * NFP + f] = a2 * sn;
  }
}

// ---------------------------------------------------------------------------
// Kernel 3: fused conv1d(k=5) GEMM + bias + LayerNorm, one block per (b,t).
//   Y[128, 257] = W[128,100] @ im2col[100, 257]   via v_wmma_f32_16x16x4_f32
//
// LDS B panel is stored K-pair-interleaved: element (k, f) lives at float
// index (k>>1)*544 + 2*f + (k&1). A lane's B fragment (K = k0+2*half, K+1 at
// column f) is then ONE aligned ds_load_b64; the k/5, k%5 im2col math happens
// once in the fill loop, not in the 225 hot iterations.
//
// Waves are mapped 2-D: mt = wv&7 fixed per wave, so the 25 A fragments are
// preloaded to registers once and reused across all 9 N-tiles.
// Hot loop = 1 ds_load_b64 + 1 v_wmma per K-step.
// ---------------------------------------------------------------------------
__global__ __launch_bounds__(512, 1) void conv_ln_kernel(
    const float* __restrict__ xs, const float* __restrict__ w,
    const float* __restrict__ bias, const float* __restrict__ lnw,
    const float* __restrict__ lnb, float* __restrict__ out) {
  __shared__ float w_lds[NDM * NK];     // 12800 floats (51.2 KB)
  __shared__ float xldB[NK * NFP];      // 27200 floats (108.8 KB), pair-interleaved
  __shared__ float cb[NDM];
  __shared__ float s_red[2];
  __shared__ float s_mv[2];

  const int tid = threadIdx.x;
  const int bt  = blockIdx.x;
  const int b   = bt / NT, t = bt % NT;

  for (int i = tid; i < NDM * NK; i += 512) w_lds[i] = w[i];
  if (tid < NDM) cb[tid] = bias[tid];
  if (tid == 0) { s_red[0] = 0.f; s_red[1] = 0.f; }
  // fill pair-interleaved B panel (im2col + SAME-pad halo resolved here)
  for (int d = tid; d < NK * NFP; d += 512) {
    const int p  = d / PAIRBLK;          // K pair
    const int r  = d - p * PAIRBLK;
    const int f  = r >> 1;
    const int k  = 2 * p + (r & 1);      // im2col K = c*5 + kk
    const int c  = k / 5;
    const int kk = k - 5 * c;
    const int tt = t + kk - 2;
    float v = 0.f;
    if (tt >= 0 && tt < NT) v = xs[(size_t)(b * NT + tt) * ROWF + c * NFP + f];
    xldB[d] = v;
  }
  __syncthreads();

  // scalar wave id -> all tile branches are s_cbranch, EXEC all-1 at WMMA
  const int wv   = __builtin_amdgcn_readfirstlane((int)(threadIdx.x >> 5));
  const int mt   = wv & 7;          // M tile: fixed per wave -> A reuse
  const int col  = wv >> 3;         // N tile column (0: nt 0..8, 1: nt 9..16)
  const int lane = tid & 31;
  const int half = lane >> 4;       // K-group within fragment
  const int lq   = lane & 15;       // row (A) / col (B,D low) index

  // preload the wave's 25 A fragments (W rows mt*16+lq, K = 4j + 2*half)
  const float* wrow = &w_lds[(mt * 16 + lq) * NK + 2 * half];
  v2f af[KSTEPS];
  #pragma unroll
  for (int j = 0; j < KSTEPS; ++j) af[j] = *(const v2f*)(wrow + 4 * j);

  v8f acc[TPW];
  float lsum = 0.f, lsq = 0.f;

  #pragma unroll
  for (int i = 0; i < TPW; ++i) {
    const int nt = col * TPW + i;
    v8f c = {0.f, 0.f, 0.f, 0.f, 0.f, 0.f, 0.f, 0.f};
    if (nt < NNT) {
      const int f = nt * 16 + lq;
      const float* bp = &xldB[half * PAIRBLK + 2 * f];
      #pragma unroll
      for (int j = 0; j < KSTEPS; ++j) {
        const v2f bf = *(const v2f*)(bp + j * (2 * PAIRBLK));  // one b64
        c = __builtin_amdgcn_wmma_f32_16x16x4_f32(
                /*neg_a=*/false, af[j], /*neg_b=*/false, bf,
                /*c_mod=*/(short)0, c, /*reuse_a=*/false, /*reuse_b=*/false);
      }
      const bool fv = (f < NF);
      #pragma unroll
      for (int r = 0; r < 8; ++r) {
        c[r] += cb[mt * 16 + r + 8 * half];
        if (fv) { lsum += c[r]; lsq += c[r] * c[r]; }
      }
    }
    acc[i] = c;
  }

  // block-wide sum / sumsq for LayerNorm over (d_model, F)
  #pragma unroll
  for (int m = 16; m > 0; m >>= 1) {
    lsum += __shfl_xor(lsum, m, 32);
    lsq  += __shfl_xor(lsq,  m, 32);
  }
  if (lane == 0) { atomicAdd(&s_red[0], lsum); atomicAdd(&s_red[1], lsq); }
  __syncthreads();
  if (tid == 0) {
    const float invN = 1.0f / (float)(NDM * NF);
    const float mu  = s_red[0] * invN;
    const float var = s_red[1] * invN - mu * mu;
    s_mv[0] = mu;
    s_mv[1] = rsqrtf(var + 1e-5f);
  }
  __syncthreads();
  const float mu = s_mv[0], rstd = s_mv[1];

  #pragma unroll
  for (int i = 0; i < TPW; ++i) {
    const int nt = col * TPW + i;
    if (nt < NNT) {
      const int f = nt * 16 + lq;
      if (f < NF) {
        #pragma unroll
        for (int r = 0; r < 8; ++r) {
          const int d = mt * 16 + r + 8 * half;
          const float g  = lnw[d * NF + f];
          const float be = lnb[d * NF + f];
          out[((size_t)bt * NDM + d) * NF + f] = (acc[i][r] - mu) * rstd * g + be;
        }
      }
    }
  }
}

// ---------------------------------------------------------------------------
extern "C" void kernel_launch(void* const* d_in, const int* in_sizes, int n_in,
                              void* d_out, int out_size, void* d_ws, size_t ws_size,
                              hipStream_t stream) {
  (void)in_sizes; (void)n_in; (void)out_size; (void)ws_size;
  const float* x    = (const float*)d_in[0];
  const float* expo = (const float*)d_in[1];
  const float* ipd  = (const float*)d_in[2];
  const float* cw   = (const float*)d_in[3];
  const float* cb   = (const float*)d_in[4];
  const float* lnw  = (const float*)d_in[5];
  const float* lnb  = (const float*)d_in[6];
  float* out = (float*)d_out;

  float* stats = (float*)d_ws;              // B*F*12 floats (~49 KB)
  float* xs    = stats + 16384;             // B*T*20*272 floats (~44.6 MB)

  stats_kernel<<<NB * NF, 128, 0, stream>>>(x, stats);
  feat_kernel<<<NB * NT, 288, 0, stream>>>(x, stats, expo, ipd, xs);
  conv_ln_kernel<<<NB * NT, 512, 0, stream>>>(xs, cw, cb, lnw, lnb, out);
}